// FastAssociativeMemory_70068096467216
// MI455X (gfx1250) — compile-verified
//
#include <hip/hip_runtime.h>
#include <math.h>

typedef __attribute__((ext_vector_type(2))) float v2f;
typedef __attribute__((ext_vector_type(8))) float v8f;

#define D_DIM 512
#define V_DIM 512
#define C_DIM 2048
#define TOPK 32
#define EPSF 1e-8f

// ---------------- Phase A: q = tanh(qk @ W_k^T + b_k) ----------------
// One wave per output row (512 rows). grid=64, block=256 (8 waves).
__global__ void compute_q_kernel(const float* __restrict__ qk,
                                 const float* __restrict__ Wk,
                                 const float* __restrict__ bk,
                                 float* __restrict__ q_out) {
    const int lane = threadIdx.x & 31;
    const int wave = threadIdx.x >> 5;
    const int row  = blockIdx.x * 8 + wave;
    const float* wrow = Wk + (size_t)row * D_DIM;
    float s = 0.0f;
#pragma unroll
    for (int t = 0; t < D_DIM / 32; ++t) {
        int j = lane + 32 * t;
        s += wrow[j] * qk[j];
    }
#pragma unroll
    for (int m = 16; m >= 1; m >>= 1) s += __shfl_xor(s, m, 32);
    if (lane == 0) q_out[row] = tanhf(s + bk[row]);
}

// ---------------- Phase B: sims_unscaled[c] = (keys_c . q) / ||keys_c|| ----
// One wave per slot (2048 slots). grid=256, block=256.
__global__ void sims_kernel(const float* __restrict__ assoc,
                            const float* __restrict__ q,
                            float* __restrict__ sims) {
    const int lane = threadIdx.x & 31;
    const int wave = threadIdx.x >> 5;
    const int c    = blockIdx.x * 8 + wave;
    const float* base = assoc + (size_t)c * D_DIM * V_DIM;  // column 0 elems at d*V_DIM
    float dot = 0.0f, nrm = 0.0f;
#pragma unroll
    for (int t = 0; t < D_DIM / 32; ++t) {
        int d = lane + 32 * t;
        float a = base[(size_t)d * V_DIM];
        dot += a * q[d];
        nrm += a * a;
    }
#pragma unroll
    for (int m = 16; m >= 1; m >>= 1) {
        dot += __shfl_xor(dot, m, 32);
        nrm += __shfl_xor(nrm, m, 32);
    }
    if (lane == 0) {
        float kn = fmaxf(sqrtf(nrm), EPSF);
        sims[c] = dot / kn;
    }
}

// ---------------- Phase C: top-32 (sorted desc, ties -> lowest idx) --------
// Single block of 256 threads. Also computes q_norm and confidences.
__global__ void topk_kernel(const float* __restrict__ q,
                            const float* __restrict__ sims,
                            const float* __restrict__ usage,
                            float* __restrict__ out,   // d_out base
                            int* __restrict__ top_idx) {
    __shared__ float s[C_DIM];
    __shared__ float redv[256];
    __shared__ int   redi[256];
    __shared__ float qn_sh;
    const int t = threadIdx.x;

    // q_norm
    float qs = 0.0f;
#pragma unroll
    for (int i = 0; i < D_DIM / 256; ++i) {
        float v = q[t + 256 * i];
        qs += v * v;
    }
    redv[t] = qs;
    __syncthreads();
    for (int off = 128; off >= 1; off >>= 1) {
        if (t < off) redv[t] += redv[t + off];
        __syncthreads();
    }
    if (t == 0) qn_sh = fmaxf(sqrtf(redv[0]), EPSF);

    // stage sims in LDS
#pragma unroll
    for (int i = 0; i < C_DIM / 256; ++i) s[t + 256 * i] = sims[t + 256 * i];
    __syncthreads();

    const float qn = qn_sh;
    for (int k = 0; k < TOPK; ++k) {
        float best = -INFINITY;
        int bidx = C_DIM;
#pragma unroll
        for (int i = 0; i < C_DIM / 256; ++i) {
            int idx = t + 256 * i;
            float v = s[idx];
            if (v > best) { best = v; bidx = idx; }   // ascending scan keeps lowest idx on tie
        }
        redv[t] = best;
        redi[t] = bidx;
        __syncthreads();
        for (int off = 128; off >= 1; off >>= 1) {
            if (t < off) {
                float vo = redv[t + off];
                int io = redi[t + off];
                if (vo > redv[t] || (vo == redv[t] && io < redi[t])) {
                    redv[t] = vo; redi[t] = io;
                }
            }
            __syncthreads();
        }
        if (t == 0) {
            int idx = redi[0];
            float sim = redv[0] / qn;
            top_idx[k] = idx;
            out[TOPK * V_DIM + k]        = sim * (1.0f + log1pf(usage[idx])); // confidences
            out[TOPK * V_DIM + TOPK + k] = sim;                                // top_sims
            s[idx] = -INFINITY;
        }
        __syncthreads();
    }
}

// ---------------- Phase D: retrieved[k,v] = sum_d q[d]*assoc[idx_k,d,v] ----
// WMMA f32 16x16x4: A = q[d..d+3] broadcast to 16 rows, B = 4x16 tile of M.
// grid = 32 slots * 4 column-blocks = 128 blocks; block = 256 (8 waves),
// each wave owns 16 consecutive output columns.
__global__ void einsum_wmma_kernel(const float* __restrict__ assoc,
                                   const float* __restrict__ q,
                                   const int* __restrict__ top_idx,
                                   float* __restrict__ out) {
    __shared__ float qs[D_DIM];
    const int t = threadIdx.x;
#pragma unroll
    for (int i = 0; i < D_DIM / 256; ++i) qs[t + 256 * i] = q[t + 256 * i];
    __syncthreads();

    const int slotK    = blockIdx.x >> 2;          // 0..31
    const int colBlock = blockIdx.x & 3;           // 0..3
    const int wave = t >> 5;
    const int lane = t & 31;
    const int n  = lane & 15;
    const int hi = lane >> 4;                      // 0: K={0,1}, 1: K={2,3}
    const int ka0 = hi * 2;
    const int ka1 = hi * 2 + 1;
    const int col = colBlock * 128 + wave * 16 + n;

    const int sel = top_idx[slotK];
    const float* Mp = assoc + (size_t)sel * D_DIM * V_DIM + col;

    v8f acc = {0.f, 0.f, 0.f, 0.f, 0.f, 0.f, 0.f, 0.f};
#pragma unroll 4
    for (int d = 0; d < D_DIM; d += 4) {
        __builtin_prefetch(&Mp[(size_t)(d + 16) * V_DIM], 0, 1);
        v2f a, b;
        a.x = qs[d + ka0];
        a.y = qs[d + ka1];
        b.x = Mp[(size_t)(d + ka0) * V_DIM];
        b.y = Mp[(size_t)(d + ka1) * V_DIM];
        acc = __builtin_amdgcn_wmma_f32_16x16x4_f32(
            /*neg_a=*/false, a, /*neg_b=*/false, b,
            /*c_mod=*/(short)0, acc, /*reuse_a=*/false, /*reuse_b=*/false);
    }
    // All 16 D-rows are identical; row 0 lives in acc[0] of lanes 0-15.
    if (hi == 0) out[(size_t)slotK * V_DIM + col] = acc[0];
}

extern "C" void kernel_launch(void* const* d_in, const int* in_sizes, int n_in,
                              void* d_out, int out_size, void* d_ws, size_t ws_size,
                              hipStream_t stream) {
    const float* qk    = (const float*)d_in[0];
    const float* Wk    = (const float*)d_in[1];
    const float* bk    = (const float*)d_in[2];
    const float* assoc = (const float*)d_in[3];
    const float* usage = (const float*)d_in[4];
    float* out = (float*)d_out;

    float* ws_q    = (float*)d_ws;          // 512 floats
    float* ws_sims = ws_q + D_DIM;          // 2048 floats
    int*   ws_idx  = (int*)(ws_sims + C_DIM); // 32 ints

    compute_q_kernel<<<D_DIM / 8, 256, 0, stream>>>(qk, Wk, bk, ws_q);
    sims_kernel<<<C_DIM / 8, 256, 0, stream>>>(assoc, ws_q, ws_sims);
    topk_kernel<<<1, 256, 0, stream>>>(ws_q, ws_sims, usage, out, ws_idx);
    einsum_wmma_kernel<<<TOPK * 4, 256, 0, stream>>>(assoc, ws_q, ws_idx, out);
}